// SFDiVeQDetach_78426102825291
// MI455X (gfx1250) — compile-verified
//
#include <hip/hip_runtime.h>
#include <hip/hip_bf16.h>
#include <math.h>

typedef __attribute__((ext_vector_type(2))) float v2f;
typedef __attribute__((ext_vector_type(8))) float v8f;

#define DIMS 64
#define WAVES_PER_BLOCK 8
#define ROWS_PER_BLOCK 128      // 8 waves x 16 rows
#define STAGE_CODES 128         // dith codes staged in LDS per stage
#define ROW_STRIDE 68           // padded LDS row stride (floats): bank stride 4 -> conflict-free
#define BIG_NORM 3.0e38f
#define EPSV 1e-8f

// ---------------------------------------------------------------------------
// Phase 0: dith[k] = (1-lam_k)*cb[k] + lam_k*cb[k+1], norms, pad row, zero loss
// grid = KP blocks, block = 32 (one wave per code row)
// ---------------------------------------------------------------------------
__global__ __launch_bounds__(32) void vq_phase0(const float* __restrict__ cb,
                                                const float* __restrict__ lambda_vals,
                                                float* __restrict__ dith,
                                                float* __restrict__ norms,
                                                float* __restrict__ loss_slot,
                                                int K) {
    const int k = blockIdx.x;
    const int lane = threadIdx.x;
    if (k == 0 && lane == 0) *loss_slot = 0.0f;

    const int d = lane * 2;
    float v0 = 0.0f, v1 = 0.0f;
    if (k < K) {
        const float lam = lambda_vals[k];
        const float* ci  = cb + (size_t)k * DIMS;
        const float* cip = ci + DIMS;
        v0 = (1.0f - lam) * ci[d]     + lam * cip[d];
        v1 = (1.0f - lam) * ci[d + 1] + lam * cip[d + 1];
    }
    dith[(size_t)k * DIMS + d]     = v0;
    dith[(size_t)k * DIMS + d + 1] = v1;

    float s = v0 * v0 + v1 * v1;
    #pragma unroll
    for (int m = 1; m <= 16; m <<= 1) s += __shfl_xor(s, m, 32);
    if (lane == 0) norms[k] = (k < K) ? s : BIG_NORM;
}

// ---------------------------------------------------------------------------
// Phase 1: argmin_k ( ||dith_k||^2 - 2 * z_m . dith_k )  via f32 WMMA
// grid = N/128 blocks, block = 256 (8 waves). Wave w owns rows base+w*16..+15.
// Four 16-code subtiles in flight -> same-acc WMMA dependency distance = 4.
// ---------------------------------------------------------------------------
__global__ __launch_bounds__(256) void vq_phase1(const float* __restrict__ z,
                                                 const float* __restrict__ dith,
                                                 const float* __restrict__ norms,
                                                 float* __restrict__ idx_out,
                                                 int nCodesPadded) {
    __shared__ float sB[STAGE_CODES * ROW_STRIDE];
    __shared__ float sN[STAGE_CODES];

    const int tid  = threadIdx.x;
    const int lane = tid & 31;
    const int wave = tid >> 5;
    const int half = lane >> 4;   // 0: K-lanes {0,1}, 1: K-lanes {2,3}
    const int l16  = lane & 15;

    const int rowBase = blockIdx.x * ROWS_PER_BLOCK + wave * 16;
    const int myRow   = rowBase + l16;

    // A tile resident in registers: 16 chunks (K=4 each) of the 16x64 row block.
    // ISA 16x4 f32 A layout: lane<16 -> (K=0,K=1), lane>=16 -> (K=2,K=3).
    v2f a[16];
    {
        const float* zrow = z + (size_t)myRow * DIMS + 2 * half;
        #pragma unroll
        for (int c = 0; c < 16; ++c)
            a[c] = *(const v2f*)(zrow + 4 * c);
    }

    float vmin[8];
    int   vidx[8];
    #pragma unroll
    for (int e = 0; e < 8; ++e) { vmin[e] = BIG_NORM; vidx[e] = 0; }

    const int nStages = nCodesPadded / STAGE_CODES;
    for (int s = 0; s < nStages; ++s) {
        __syncthreads();
        // cooperative stage of 128 codes x 64 dims into padded LDS
        for (int i = tid; i < STAGE_CODES * (DIMS / 4); i += 256) {
            const int r  = i >> 4;          // code row in stage
            const int d4 = (i & 15) * 4;    // dim offset
            const float4 v = *(const float4*)(dith + ((size_t)(s * STAGE_CODES + r)) * DIMS + d4);
            float* dst = &sB[r * ROW_STRIDE + d4];
            dst[0] = v.x; dst[1] = v.y; dst[2] = v.z; dst[3] = v.w;
        }
        if (tid < STAGE_CODES) sN[tid] = norms[s * STAGE_CODES + tid];
        __syncthreads();

        // prefetch next stage of dith into cache while we compute
        if (s + 1 < nStages)
            __builtin_prefetch(dith + (size_t)(s + 1) * STAGE_CODES * DIMS + tid * 32, 0, 0);

        // 8 subtiles of 16 codes, FOUR in flight for WMMA pipe latency hiding
        #pragma unroll
        for (int sub = 0; sub < STAGE_CODES / 16; sub += 4) {
            v8f acc0 = {};
            v8f acc1 = {};
            v8f acc2 = {};
            v8f acc3 = {};
            const float* b0 = &sB[((sub + 0) * 16 + l16) * ROW_STRIDE + 2 * half];
            const float* b1 = &sB[((sub + 1) * 16 + l16) * ROW_STRIDE + 2 * half];
            const float* b2 = &sB[((sub + 2) * 16 + l16) * ROW_STRIDE + 2 * half];
            const float* b3 = &sB[((sub + 3) * 16 + l16) * ROW_STRIDE + 2 * half];
            #pragma unroll
            for (int c = 0; c < 16; ++c) {
                const v2f bf0 = *(const v2f*)(b0 + 4 * c);
                const v2f bf1 = *(const v2f*)(b1 + 4 * c);
                const v2f bf2 = *(const v2f*)(b2 + 4 * c);
                const v2f bf3 = *(const v2f*)(b3 + 4 * c);
                acc0 = __builtin_amdgcn_wmma_f32_16x16x4_f32(
                    false, a[c], false, bf0, (short)0, acc0, false, false);
                acc1 = __builtin_amdgcn_wmma_f32_16x16x4_f32(
                    false, a[c], false, bf1, (short)0, acc1, false, false);
                acc2 = __builtin_amdgcn_wmma_f32_16x16x4_f32(
                    false, a[c], false, bf2, (short)0, acc2, false, false);
                acc3 = __builtin_amdgcn_wmma_f32_16x16x4_f32(
                    false, a[c], false, bf3, (short)0, acc3, false, false);
            }
            const float n0 = sN[(sub + 0) * 16 + l16];
            const float n1 = sN[(sub + 1) * 16 + l16];
            const float n2 = sN[(sub + 2) * 16 + l16];
            const float n3 = sN[(sub + 3) * 16 + l16];
            const int code0 = s * STAGE_CODES + sub * 16 + l16;
            #pragma unroll
            for (int e = 0; e < 8; ++e) {
                const float s0 = n0 - 2.0f * acc0[e];
                const float s1 = n1 - 2.0f * acc1[e];
                const float s2 = n2 - 2.0f * acc2[e];
                const float s3 = n3 - 2.0f * acc3[e];
                if (s0 < vmin[e]) { vmin[e] = s0; vidx[e] = code0; }
                if (s1 < vmin[e]) { vmin[e] = s1; vidx[e] = code0 + 16; }
                if (s2 < vmin[e]) { vmin[e] = s2; vidx[e] = code0 + 32; }
                if (s3 < vmin[e]) { vmin[e] = s3; vidx[e] = code0 + 48; }
            }
        }
    }

    // C layout: element e on this lane is (row = e + 8*half, col = l16).
    // Reduce (min,argmin) across the 16 lanes of each half (xor masks < 16).
    #pragma unroll
    for (int e = 0; e < 8; ++e) {
        float v = vmin[e];
        int   ix = vidx[e];
        #pragma unroll
        for (int m = 1; m <= 8; m <<= 1) {
            const float ov = __shfl_xor(v, m, 32);
            const int   oi = __shfl_xor(ix, m, 32);
            if (ov < v || (ov == v && oi < ix)) { v = ov; ix = oi; }
        }
        if (l16 == 0)
            idx_out[rowBase + e + 8 * half] = (float)ix;   // exact for ix < 2^24
    }
}

// ---------------------------------------------------------------------------
// Phase 2: straight-through z_q + loss. One wave per row; lane handles 2 dims.
// grid = N/8 blocks, block = 256 (8 waves).
// ---------------------------------------------------------------------------
__global__ __launch_bounds__(256) void vq_phase2(const float* __restrict__ z,
                                                 const float* __restrict__ cb,
                                                 const float* __restrict__ lambda_vals,
                                                 const float* __restrict__ idx_f,
                                                 float* __restrict__ zq,
                                                 float* __restrict__ loss_slot,
                                                 int nRows) {
    const int wave = threadIdx.x >> 5;
    const int lane = threadIdx.x & 31;
    const int row  = blockIdx.x * WAVES_PER_BLOCK + wave;

    const int idx = (int)idx_f[row];
    const float lam = lambda_vals[idx];
    const float* zr  = z  + (size_t)row * DIMS;
    const float* ci  = cb + (size_t)idx * DIMS;
    const float* cip = ci + DIMS;

    const int d = lane * 2;
    const float z0 = zr[d],  z1 = zr[d + 1];
    const float a0 = ci[d],  a1 = ci[d + 1];
    const float b0 = cip[d], b1 = cip[d + 1];

    const float di0 = a0 - z0, di1 = a1 - z1;
    const float dp0 = b0 - z0, dp1 = b1 - z1;
    const float t0 = (1.0f - lam) * a0 + lam * b0 - z0;   // dith_t - z
    const float t1 = (1.0f - lam) * a1 + lam * b1 - z1;

    float si = di0 * di0 + di1 * di1;
    float sp = dp0 * dp0 + dp1 * dp1;
    float sl = t0 * t0 + t1 * t1;
    #pragma unroll
    for (int m = 1; m <= 16; m <<= 1) {
        si += __shfl_xor(si, m, 32);
        sp += __shfl_xor(sp, m, 32);
        sl += __shfl_xor(sl, m, 32);
    }

    const float mi = sqrtf(si);
    const float mp = sqrtf(sp);
    const float wi = mi * (1.0f - lam) / (mi + EPSV);
    const float wp = mp * lam / (mp + EPSV);

    float* q = zq + (size_t)row * DIMS;
    q[d]     = z0 + wi * di0 + wp * dp0;
    q[d + 1] = z1 + wi * di1 + wp * dp1;

    if (lane == 0) {
        // loss = codebook_loss + 0.25*commitment = 1.25 * mean((dith_t - z)^2)
        atomicAdd(loss_slot, 1.25f * sl / (float)((size_t)nRows * DIMS));
    }
}

// ---------------------------------------------------------------------------
extern "C" void kernel_launch(void* const* d_in, const int* in_sizes, int n_in,
                              void* d_out, int out_size, void* d_ws, size_t ws_size,
                              hipStream_t stream) {
    const float* z   = (const float*)d_in[0];
    const float* cb  = (const float*)d_in[1];
    const float* lam = (const float*)d_in[2];

    const int zElems = in_sizes[0];            // 8388608
    const int K      = in_sizes[2];            // 1023
    const int N      = zElems / DIMS;          // 131072
    const int KP     = ((K + STAGE_CODES - 1) / STAGE_CODES) * STAGE_CODES; // 1024

    float* out   = (float*)d_out;
    float* zq    = out;                        // [N*64]
    float* loss  = out + zElems;               // [1]
    float* idxf  = out + zElems + 1;           // [N]

    float* dith  = (float*)d_ws;               // [KP*64]
    float* norms = dith + (size_t)KP * DIMS;   // [KP]

    vq_phase0<<<KP, 32, 0, stream>>>(cb, lam, dith, norms, loss, K);
    vq_phase1<<<N / ROWS_PER_BLOCK, 256, 0, stream>>>(z, dith, norms, idxf, KP);
    vq_phase2<<<N / WAVES_PER_BLOCK, 256, 0, stream>>>(z, cb, lam, idxf, zq, loss, N);
}